// GetNextIterX_84636625535603
// MI455X (gfx1250) — compile-verified
//
#include <hip/hip_runtime.h>

// ---------------------------------------------------------------------------
// Frequency-domain ADMM X-update (convolutional sparse coding) for MI455X.
// FFTs are done as radix-16 four-step transforms where each 16-point DFT
// stage is a complex 16x16x16 matmul executed with V_WMMA_F32_16X16X4_F32
// (4 chained K=4 WMMAs per real matmul, 16 per complex matmul, 32 per FFT).
// One wave32 owns one 256-point FFT; data staged via LDS tiles laid out to
// match the CDNA5 WMMA A / C/D VGPR mappings.
// Column-FFT gather loads use GLOBAL_LOAD_ASYNC_TO_LDS_B32 (ASYNCcnt path):
// global -> LDS with no VGPR round trip, waited with s_wait_asynccnt.
// Workspace usage (floats): Dhat 12,681,216 | Ginv 594,432 | Zf 1,585,152 |
// Af 4,227,072  => ~76.3 MB total (stays resident in the 192 MB L2).
// ---------------------------------------------------------------------------

#define TWO_PI 6.28318530717958647692f
#define RHO_C  1.0f

#define Bdim  8
#define Hdim  256
#define Wdim  256
#define Cdim  3
#define Mdim  64
#define WFdim 129
#define NF    (Hdim * WFdim)   // 33024 frequencies

typedef float v2f __attribute__((ext_vector_type(2)));
typedef float v8f __attribute__((ext_vector_type(8)));

struct cplx { float re, im; };
__device__ __forceinline__ cplx cmul (cplx a, cplx b) { return {a.re*b.re - a.im*b.im, a.re*b.im + a.im*b.re}; }
__device__ __forceinline__ cplx cmulc(cplx a, cplx b) { return {a.re*b.re + a.im*b.im, a.im*b.re - a.re*b.im}; } // a*conj(b)
__device__ __forceinline__ cplx cadd (cplx a, cplx b) { return {a.re + b.re, a.im + b.im}; }
__device__ __forceinline__ cplx csub (cplx a, cplx b) { return {a.re - b.re, a.im - b.im}; }
__device__ __forceinline__ cplx cdiv (cplx a, cplx b) {
  float inv = 1.0f / (b.re*b.re + b.im*b.im);
  return {(a.re*b.re + a.im*b.im)*inv, (a.im*b.re - a.re*b.im)*inv};
}
__device__ __forceinline__ cplx ldc(const float* p, size_t i) { return {p[2*i], p[2*i+1]}; }
__device__ __forceinline__ void stc(float* p, size_t i, cplx v) { p[2*i] = v.re; p[2*i+1] = v.im; }

// ---------------------------------------------------------------------------
// Async global->LDS copy (CDNA5). The generic address of a __shared__ object
// is {SHARED_BASE, lds_byte_offset[31:0]}, so the low 32 bits are exactly the
// LDS address the instruction's VDST operand wants.
// ---------------------------------------------------------------------------
__device__ __forceinline__ unsigned lds_addr_of(const void* p) {
  return (unsigned)(uintptr_t)p;
}
__device__ __forceinline__ void async_copy_f32(unsigned lds_dst, const float* gsrc) {
  asm volatile("global_load_async_to_lds_b32 %0, %1, off"
               :: "v"(lds_dst), "v"((unsigned long long)(uintptr_t)gsrc)
               : "memory");
}
__device__ __forceinline__ void async_wait0() {
  asm volatile("s_wait_asynccnt 0x0" ::: "memory");
}

// ---------------------------------------------------------------------------
// One real 16x16x16 matmul accumulated into acc via 4 chained K=4 fp32 WMMAs.
// A tile: A[m][k] at A[m*17+k] (LDS, padded row stride 17).
// B tile: B[k][n] at B[k*17+n].
// ISA layouts: A 16x4 -> lane m=lane%16, VGPR0 K={0|2}, VGPR1 K={1|3} split
// by half-wave; B 4x16 mirrored; C/D: VGPR v -> row v+8*(lane>=16), col lane%16.
// ---------------------------------------------------------------------------
__device__ __forceinline__ v8f cmatmac(const float* A, const float* B, v8f acc, float ascale) {
  const int lane = threadIdx.x & 31;
  const int hi   = lane >> 4;
  const int lo   = lane & 15;
#pragma unroll
  for (int kb = 0; kb < 16; kb += 4) {
    const int ka = kb + 2 * hi;
    v2f a, b;
    a.x = ascale * A[lo * 17 + ka];
    a.y = ascale * A[lo * 17 + ka + 1];
    b.x = B[ka * 17 + lo];
    b.y = B[(ka + 1) * 17 + lo];
    acc = __builtin_amdgcn_wmma_f32_16x16x4_f32(false, a, false, b, (short)0, acc, false, false);
  }
  return acc;
}

// ---------------------------------------------------------------------------
// 256-point complex DFT (sgn = -1 forward, +1 inverse, unscaled) on one wave.
// x stored as tile[n>>4][n&15] with row stride 17.  Four-step:
//   A = F16 * Xmat ; A[k1][n2] *= cis(sgn*2pi*k1*n2/256) ; C = A * F16
//   out[k1 + 16*k2] = C[k1][k2]
// ---------------------------------------------------------------------------
__device__ void fft256(float* xr, float* xi, const float* Fr, const float* Fi, float sgn) {
  const int lane = threadIdx.x & 31;
  const int hi   = lane >> 4;
  const int lo   = lane & 15;

  // Stage 1: complex C = F * X  (contract over n1 = row index of X)
  v8f cr = {}, ci = {};
  cr = cmatmac(Fr, xr, cr,  1.0f);
  cr = cmatmac(Fi, xi, cr, -1.0f);
  ci = cmatmac(Fr, xi, ci,  1.0f);
  ci = cmatmac(Fi, xr, ci,  1.0f);
  __syncthreads();
#pragma unroll
  for (int v = 0; v < 8; ++v) {
    const int r = v + 8 * hi;
    xr[r * 17 + lo] = cr[v];
    xi[r * 17 + lo] = ci[v];
  }
  __syncthreads();

  // Twiddle: element [k1][n2] *= cis(sgn * 2pi * k1*n2 / 256)
#pragma unroll
  for (int e = 0; e < 8; ++e) {
    const int idx = e * 32 + lane;
    const int k1 = idx >> 4, n2 = idx & 15;
    float s, c;
    __sincosf(sgn * (TWO_PI / 256.0f) * (float)(k1 * n2), &s, &c);
    const float ar = xr[k1 * 17 + n2], ai = xi[k1 * 17 + n2];
    xr[k1 * 17 + n2] = ar * c - ai * s;
    xi[k1 * 17 + n2] = ar * s + ai * c;
  }
  __syncthreads();

  // Stage 2: complex C2 = T * F  (contract over n2)
  v8f dr = {}, di = {};
  dr = cmatmac(xr, Fr, dr,  1.0f);
  dr = cmatmac(xi, Fi, dr, -1.0f);
  di = cmatmac(xr, Fi, di,  1.0f);
  di = cmatmac(xi, Fr, di,  1.0f);
  __syncthreads();
  // C2 row = k1 = v+8*hi, col = k2 = lo ; output index o = k1 + 16*k2
  // store at tile[o>>4][o&15] = tile[k2][k1]
#pragma unroll
  for (int v = 0; v < 8; ++v) {
    const int k1 = v + 8 * hi;
    xr[lo * 17 + k1] = dr[v];
    xi[lo * 17 + k1] = di[v];
  }
  __syncthreads();
}

#define FFT_SHARED                                       \
  __shared__ float Fr[16 * 17], Fi[16 * 17];             \
  __shared__ float XR[8][16 * 17], XI[8][16 * 17];

#define FFT_BUILD_F(SGN)                                                   \
  {                                                                        \
    const int k_ = threadIdx.x >> 4, n_ = threadIdx.x & 15;                \
    float s_, c_;                                                          \
    __sincosf((SGN) * (TWO_PI / 16.0f) * (float)(k_ * n_), &s_, &c_);      \
    Fr[k_ * 17 + n_] = c_;                                                 \
    Fi[k_ * 17 + n_] = s_;                                                 \
  }                                                                        \
  __syncthreads();

// ---------------------------------------------------------------------------
// Kernel 1: Dhat[h][wf][c][m] = sum_{kh,kw} D[kh][kw][c][m] cis(-2pi(h kh + wf kw)/256)
// ---------------------------------------------------------------------------
__global__ __launch_bounds__(256) void kDhat(const float* __restrict__ D_sp,
                                             float* __restrict__ Dhat) {
  const size_t i = (size_t)blockIdx.x * 256 + threadIdx.x;
  if (i >= (size_t)NF * Cdim * Mdim) return;
  const int m = (int)(i % Mdim);
  size_t t = i / Mdim;
  const int c  = (int)(t % Cdim); t /= Cdim;
  const int wf = (int)(t % WFdim);
  const int h  = (int)(t / WFdim);
  float sr = 0.0f, si = 0.0f;
  for (int kh = 0; kh < 11; ++kh) {
    for (int kw = 0; kw < 11; ++kw) {
      const float v = D_sp[(((size_t)kh * 11 + kw) * Cdim + c) * Mdim + m];
      const int ph = (h * kh + wf * kw) & 255;  // phase mod 256 for accuracy
      float s, co;
      __sincosf(-(TWO_PI / 256.0f) * (float)ph, &s, &co);
      sr += v * co;
      si += v * s;
    }
  }
  Dhat[2 * i]     = sr;
  Dhat[2 * i + 1] = si;
}

// ---------------------------------------------------------------------------
// Kernel 2: Ginv[f] = (Dhat_f Dhat_f^H + rho I_3)^{-1}   (3x3 complex, adjugate)
// ---------------------------------------------------------------------------
__global__ __launch_bounds__(256) void kGinv(const float* __restrict__ Dhat,
                                             float* __restrict__ Ginv) {
  const int f = blockIdx.x * 256 + threadIdx.x;
  if (f >= NF) return;
  cplx g[3][3];
  for (int r = 0; r < 3; ++r)
    for (int k = 0; k < 3; ++k) g[r][k] = {0.0f, 0.0f};
  for (int m = 0; m < Mdim; ++m) {
    cplx d[3];
    for (int c = 0; c < 3; ++c) d[c] = ldc(Dhat, ((size_t)f * Cdim + c) * Mdim + m);
    for (int r = 0; r < 3; ++r)
      for (int k = 0; k < 3; ++k) g[r][k] = cadd(g[r][k], cmulc(d[r], d[k]));
  }
  g[0][0].re += RHO_C; g[1][1].re += RHO_C; g[2][2].re += RHO_C;

  cplx A00 = csub(cmul(g[1][1], g[2][2]), cmul(g[1][2], g[2][1]));
  cplx A01 = csub(cmul(g[0][2], g[2][1]), cmul(g[0][1], g[2][2]));
  cplx A02 = csub(cmul(g[0][1], g[1][2]), cmul(g[0][2], g[1][1]));
  cplx A10 = csub(cmul(g[1][2], g[2][0]), cmul(g[1][0], g[2][2]));
  cplx A11 = csub(cmul(g[0][0], g[2][2]), cmul(g[0][2], g[2][0]));
  cplx A12 = csub(cmul(g[0][2], g[1][0]), cmul(g[0][0], g[1][2]));
  cplx A20 = csub(cmul(g[1][0], g[2][1]), cmul(g[1][1], g[2][0]));
  cplx A21 = csub(cmul(g[0][1], g[2][0]), cmul(g[0][0], g[2][1]));
  cplx A22 = csub(cmul(g[0][0], g[1][1]), cmul(g[0][1], g[1][0]));
  cplx det = cadd(cadd(cmul(g[0][0], A00), cmul(g[0][1], A10)), cmul(g[0][2], A20));

  stc(Ginv, (size_t)f * 9 + 0, cdiv(A00, det));
  stc(Ginv, (size_t)f * 9 + 1, cdiv(A01, det));
  stc(Ginv, (size_t)f * 9 + 2, cdiv(A02, det));
  stc(Ginv, (size_t)f * 9 + 3, cdiv(A10, det));
  stc(Ginv, (size_t)f * 9 + 4, cdiv(A11, det));
  stc(Ginv, (size_t)f * 9 + 5, cdiv(A12, det));
  stc(Ginv, (size_t)f * 9 + 6, cdiv(A20, det));
  stc(Ginv, (size_t)f * 9 + 7, cdiv(A21, det));
  stc(Ginv, (size_t)f * 9 + 8, cdiv(A22, det));
}

// ---------------------------------------------------------------------------
// Kernel 3: row FFTs of z_prevlayer -> Zf[b][h][wf][c]  (keep wf < 129)
// unit = (b*H + h)*C + c ; one wave per row FFT.
// ---------------------------------------------------------------------------
__global__ __launch_bounds__(256) void kRowZ(const float* __restrict__ z,
                                             float* __restrict__ Zf) {
  FFT_SHARED
  FFT_BUILD_F(-1.0f)
  const int wv = threadIdx.x >> 5, lane = threadIdx.x & 31;
  const int unit = blockIdx.x * 8 + wv;
  const int c  = unit % Cdim;
  const int bh = unit / Cdim;
  const int h  = bh % Hdim;
  const int b  = bh / Hdim;
  float* xr = XR[wv];
  float* xi = XI[wv];
#pragma unroll
  for (int e = 0; e < 8; ++e) {
    const int n = e * 32 + lane;
    xr[(n >> 4) * 17 + (n & 15)] = z[(((size_t)b * Hdim + h) * Wdim + n) * Cdim + c];
    xi[(n >> 4) * 17 + (n & 15)] = 0.0f;
  }
  __syncthreads();
  fft256(xr, xi, Fr, Fi, -1.0f);
  for (int e = 0; e < 8; ++e) {
    const int n = e * 32 + lane;
    if (n < WFdim) {
      stc(Zf, ((size_t)b * NF + (size_t)h * WFdim + n) * Cdim + c,
          {xr[(n >> 4) * 17 + (n & 15)], xi[(n >> 4) * 17 + (n & 15)]});
    }
  }
}

// ---------------------------------------------------------------------------
// Kernel 4: column FFTs (over h) of Zf, in place.  unit = (b*WF + wf)*C + c
// Gather load done with async global->LDS copies (ASYNCcnt path).
// ---------------------------------------------------------------------------
__global__ __launch_bounds__(256) void kColZ(float* __restrict__ Zf) {
  FFT_SHARED
  FFT_BUILD_F(-1.0f)
  const int wv = threadIdx.x >> 5, lane = threadIdx.x & 31;
  const int unit = blockIdx.x * 8 + wv;
  const int c  = unit % Cdim;
  const int bw = unit / Cdim;
  const int wf = bw % WFdim;
  const int b  = bw / WFdim;
  float* xr = XR[wv];
  float* xi = XI[wv];
#pragma unroll
  for (int e = 0; e < 8; ++e) {
    const int n = e * 32 + lane;  // n == h
    const size_t ci = ((size_t)b * NF + (size_t)n * WFdim + wf) * Cdim + c;
    const int slot = (n >> 4) * 17 + (n & 15);
    async_copy_f32(lds_addr_of(&xr[slot]), Zf + 2 * ci);
    async_copy_f32(lds_addr_of(&xi[slot]), Zf + 2 * ci + 1);
  }
  async_wait0();
  __syncthreads();
  fft256(xr, xi, Fr, Fi, -1.0f);
#pragma unroll
  for (int e = 0; e < 8; ++e) {
    const int n = e * 32 + lane;
    stc(Zf, ((size_t)b * NF + (size_t)n * WFdim + wf) * Cdim + c,
        {xr[(n >> 4) * 17 + (n & 15)], xi[(n >> 4) * 17 + (n & 15)]});
  }
}

// ---------------------------------------------------------------------------
// Kernel 5: row FFTs of q = rho*z_over_R + gamma for one batch b.
// unit = h*M + m ; writes Af[h][wf][m] for wf < 129.
// ---------------------------------------------------------------------------
__global__ __launch_bounds__(256) void kRowQ(const float* __restrict__ zr,
                                             const float* __restrict__ gm,
                                             float* __restrict__ Af, int b) {
  FFT_SHARED
  FFT_BUILD_F(-1.0f)
  const int wv = threadIdx.x >> 5, lane = threadIdx.x & 31;
  const int unit = blockIdx.x * 8 + wv;
  const int m = unit & (Mdim - 1);
  const int h = unit >> 6;
  float* xr = XR[wv];
  float* xi = XI[wv];
#pragma unroll
  for (int e = 0; e < 8; ++e) {
    const int n = e * 32 + lane;
    const size_t gi = (((size_t)b * Hdim + h) * Wdim + n) * Mdim + m;
    xr[(n >> 4) * 17 + (n & 15)] = RHO_C * zr[gi] + gm[gi];
    xi[(n >> 4) * 17 + (n & 15)] = 0.0f;
  }
  __syncthreads();
  fft256(xr, xi, Fr, Fi, -1.0f);
  for (int e = 0; e < 8; ++e) {
    const int n = e * 32 + lane;
    if (n < WFdim) {
      stc(Af, ((size_t)h * WFdim + n) * Mdim + m,
          {xr[(n >> 4) * 17 + (n & 15)], xi[(n >> 4) * 17 + (n & 15)]});
    }
  }
}

// ---------------------------------------------------------------------------
// Kernel 6/8: column FFT (over h) of Af, in place, direction = sgn.
// unit = wf*M + m.  Gather load via async global->LDS copies.
// ---------------------------------------------------------------------------
__global__ __launch_bounds__(256) void kColAf(float* __restrict__ Af, float sgn) {
  FFT_SHARED
  FFT_BUILD_F(sgn)
  const int wv = threadIdx.x >> 5, lane = threadIdx.x & 31;
  const int unit = blockIdx.x * 8 + wv;
  const int m  = unit & (Mdim - 1);
  const int wf = unit >> 6;
  float* xr = XR[wv];
  float* xi = XI[wv];
#pragma unroll
  for (int e = 0; e < 8; ++e) {
    const int n = e * 32 + lane;  // n == h
    const size_t ci = ((size_t)n * WFdim + wf) * Mdim + m;
    const int slot = (n >> 4) * 17 + (n & 15);
    async_copy_f32(lds_addr_of(&xr[slot]), Af + 2 * ci);
    async_copy_f32(lds_addr_of(&xi[slot]), Af + 2 * ci + 1);
  }
  async_wait0();
  __syncthreads();
  fft256(xr, xi, Fr, Fi, sgn);
#pragma unroll
  for (int e = 0; e < 8; ++e) {
    const int n = e * 32 + lane;
    stc(Af, ((size_t)n * WFdim + wf) * Mdim + m,
        {xr[(n >> 4) * 17 + (n & 15)], xi[(n >> 4) * 17 + (n & 15)]});
  }
}

// ---------------------------------------------------------------------------
// Kernel 7: per-frequency Woodbury solve.  One 64-thread block per frequency:
//   a_m = Af_m + sum_c conj(D_cm) Z_c
//   t_c = sum_m D_cm a_m            (LDS tree reduction)
//   s   = Ginv t
//   Xf_m = (a_m - sum_c conj(D_cm) s_c) / rho   -> Af (in place)
// ---------------------------------------------------------------------------
__global__ __launch_bounds__(64) void kSolve(const float* __restrict__ Dhat,
                                             const float* __restrict__ Ginv,
                                             const float* __restrict__ Zf,
                                             float* __restrict__ Af, int b) {
  __shared__ float red[6][64];
  __shared__ float sv[6];
  const int f = blockIdx.x;
  const int m = threadIdx.x;

  cplx zc[3], d[3];
#pragma unroll
  for (int c = 0; c < 3; ++c) {
    zc[c] = ldc(Zf, ((size_t)b * NF + f) * Cdim + c);
    d[c]  = ldc(Dhat, ((size_t)f * Cdim + c) * Mdim + m);
  }
  cplx a = ldc(Af, (size_t)f * Mdim + m);
#pragma unroll
  for (int c = 0; c < 3; ++c) a = cadd(a, cmulc(zc[c], d[c]));  // + conj(d)*z

#pragma unroll
  for (int c = 0; c < 3; ++c) {
    cplx t = cmul(d[c], a);
    red[2 * c][m]     = t.re;
    red[2 * c + 1][m] = t.im;
  }
  __syncthreads();
  for (int s = 32; s > 0; s >>= 1) {
    if (m < s) {
#pragma unroll
      for (int j = 0; j < 6; ++j) red[j][m] += red[j][m + s];
    }
    __syncthreads();
  }
  if (m == 0) {
    cplx t0 = {red[0][0], red[1][0]};
    cplx t1 = {red[2][0], red[3][0]};
    cplx t2 = {red[4][0], red[5][0]};
#pragma unroll
    for (int r = 0; r < 3; ++r) {
      cplx g0 = ldc(Ginv, (size_t)f * 9 + r * 3 + 0);
      cplx g1 = ldc(Ginv, (size_t)f * 9 + r * 3 + 1);
      cplx g2 = ldc(Ginv, (size_t)f * 9 + r * 3 + 2);
      cplx s  = cadd(cadd(cmul(g0, t0), cmul(g1, t1)), cmul(g2, t2));
      sv[2 * r]     = s.re;
      sv[2 * r + 1] = s.im;
    }
  }
  __syncthreads();
  cplx x = a;
#pragma unroll
  for (int c = 0; c < 3; ++c) {
    cplx s = {sv[2 * c], sv[2 * c + 1]};
    x = csub(x, cmulc(s, d[c]));  // - conj(d)*s
  }
  stc(Af, (size_t)f * Mdim + m, {x.re / RHO_C, x.im / RHO_C});
}

// ---------------------------------------------------------------------------
// Kernel 9: inverse row transform with Hermitian expansion; writes real output
// scaled by 1/(H*W).  unit = h*M + m.
// ---------------------------------------------------------------------------
__global__ __launch_bounds__(256) void kInvRow(const float* __restrict__ Af,
                                               float* __restrict__ out, int b) {
  FFT_SHARED
  FFT_BUILD_F(1.0f)
  const int wv = threadIdx.x >> 5, lane = threadIdx.x & 31;
  const int unit = blockIdx.x * 8 + wv;
  const int m = unit & (Mdim - 1);
  const int h = unit >> 6;
  float* xr = XR[wv];
  float* xi = XI[wv];
#pragma unroll
  for (int e = 0; e < 8; ++e) {
    const int n = e * 32 + lane;
    cplx v;
    if (n < WFdim) {
      v = ldc(Af, ((size_t)h * WFdim + n) * Mdim + m);
    } else {
      v = ldc(Af, ((size_t)h * WFdim + (Wdim - n)) * Mdim + m);
      v.im = -v.im;
    }
    xr[(n >> 4) * 17 + (n & 15)] = v.re;
    xi[(n >> 4) * 17 + (n & 15)] = v.im;
  }
  __syncthreads();
  fft256(xr, xi, Fr, Fi, 1.0f);
#pragma unroll
  for (int e = 0; e < 8; ++e) {
    const int w = e * 32 + lane;
    out[(((size_t)b * Hdim + h) * Wdim + w) * Mdim + m] =
        xr[(w >> 4) * 17 + (w & 15)] * (1.0f / 65536.0f);
  }
}

// ---------------------------------------------------------------------------
extern "C" void kernel_launch(void* const* d_in, const int* in_sizes, int n_in,
                              void* d_out, int out_size, void* d_ws, size_t ws_size,
                              hipStream_t stream) {
  (void)in_sizes; (void)n_in; (void)out_size; (void)ws_size;
  const float* z_prev   = (const float*)d_in[0];
  const float* z_over_R = (const float*)d_in[1];
  const float* gamma    = (const float*)d_in[2];
  const float* D_sp     = (const float*)d_in[3];
  float* out = (float*)d_out;

  float* ws   = (float*)d_ws;
  float* Dhat = ws;                                           // NF*3*64*2 = 12,681,216 floats
  float* Ginv = Dhat + (size_t)NF * Cdim * Mdim * 2;          // NF*9*2    =    594,432
  float* Zf   = Ginv + (size_t)NF * 9 * 2;                    // B*NF*3*2  =  1,585,152
  float* Af   = Zf + (size_t)Bdim * NF * Cdim * 2;            // NF*64*2   =  4,227,072

  // Precompute frequency-domain dictionary and 3x3 inverse Gram per frequency.
  kDhat<<<(NF * Cdim * Mdim) / 256, 256, 0, stream>>>(D_sp, Dhat);
  kGinv<<<(NF + 255) / 256, 256, 0, stream>>>(Dhat, Ginv);

  // rfft2 of z_prevlayer (all batches; small).
  kRowZ<<<(Bdim * Hdim * Cdim) / 8, 256, 0, stream>>>(z_prev, Zf);
  kColZ<<<(Bdim * WFdim * Cdim) / 8, 256, 0, stream>>>(Zf);

  // Per-batch pipeline (reuses the 16.9 MB Af spectrum buffer).
  for (int b = 0; b < Bdim; ++b) {
    kRowQ<<<(Hdim * Mdim) / 8, 256, 0, stream>>>(z_over_R, gamma, Af, b);
    kColAf<<<(WFdim * Mdim) / 8, 256, 0, stream>>>(Af, -1.0f);
    kSolve<<<NF, 64, 0, stream>>>(Dhat, Ginv, Zf, Af, b);
    kColAf<<<(WFdim * Mdim) / 8, 256, 0, stream>>>(Af, 1.0f);
    kInvRow<<<(Hdim * Mdim) / 8, 256, 0, stream>>>(Af, out, b);
  }
}